// Encoder_90245852823923
// MI455X (gfx1250) — compile-verified
//
#include <hip/hip_runtime.h>
#include <math.h>

typedef __attribute__((ext_vector_type(2))) float v2f;
typedef __attribute__((ext_vector_type(8))) float v8f;

#define D 128
#define G_GRAPHS 256

__device__ __forceinline__ void atomAddF(float* p, float v) {
  // agent-scope relaxed fadd -> global_atomic_add_f32 on gfx1250
  __hip_atomic_fetch_add(p, v, __ATOMIC_RELAXED, __HIP_MEMORY_SCOPE_AGENT);
}

// ---------------------------------------------------------------------------
// C[M x 128] = op(A)[M x 128] @ W[128 x 128] (+bias) (optional relu epilogue)
//   PRO:      op(A) = relu(A + abias)  applied while staging A into LDS
//   EPI_RELU: C = relu(C)
// Block: 256 threads = 8 waves. Block tile: 32 rows x 128 cols.
// Wave w: cols [16w,16w+16), two 16-row v8f accumulators.
// A tile staged in LDS (stride 132 floats -> conflict-free b64 frag loads).
// W read directly from global (each element used exactly once per block).
// M must be a multiple of 32 (100000 = 3125*32, 256 = 8*32).
// ---------------------------------------------------------------------------
template <bool PRO, bool EPI_RELU>
__global__ __launch_bounds__(256) void gemm128_wmma(
    const float* __restrict__ A, const float* __restrict__ W,
    const float* __restrict__ abias, const float* __restrict__ bias,
    float* __restrict__ C)
{
  __shared__ float As[32 * 132];
  const int tid = threadIdx.x;
  const int r0  = blockIdx.x * 32;

  // cooperative load of 32x128 A tile (1024 float4, 4 per thread),
  // with optional fused relu(A + abias) prologue
  {
    const float4* A4 = (const float4*)(A + (size_t)r0 * D);
    #pragma unroll
    for (int i = 0; i < 4; ++i) {
      int f4  = i * 256 + tid;          // 0..1023
      int row = f4 >> 5;                // f4*4 / 128
      int col = (f4 * 4) & 127;
      float4 v = A4[f4];
      if (PRO) {
        v.x = fmaxf(v.x + abias[col + 0], 0.0f);
        v.y = fmaxf(v.y + abias[col + 1], 0.0f);
        v.z = fmaxf(v.z + abias[col + 2], 0.0f);
        v.w = fmaxf(v.w + abias[col + 3], 0.0f);
      }
      *(float4*)&As[row * 132 + col] = v;
    }
  }
  __syncthreads();

  const int wave  = tid >> 5;
  const int lane  = tid & 31;
  const int half  = lane >> 4;          // 0 | 1
  const int mn    = lane & 15;          // row (A) / col (B,C) within tile
  const int n0    = wave * 16;
  const int khalf = half * 2;           // K sub-pair owned by this half-wave

  v8f acc0 = {};                        // rows r0 .. r0+15
  v8f acc1 = {};                        // rows r0+16 .. r0+31

  #pragma unroll 4
  for (int k0 = 0; k0 < D; k0 += 4) {
    // B fragment: lane holds W[k0+khalf][n0+mn], W[k0+khalf+1][n0+mn]
    const float* wp = W + (size_t)(k0 + khalf) * D + n0 + mn;
    v2f b;
    b.x = wp[0];
    b.y = wp[D];
    // A fragments: lane holds A[row][k0+khalf], A[row][k0+khalf+1]
    v2f a0 = *(const v2f*)&As[mn * 132 + k0 + khalf];
    v2f a1 = *(const v2f*)&As[(16 + mn) * 132 + k0 + khalf];
    acc0 = __builtin_amdgcn_wmma_f32_16x16x4_f32(false, a0, false, b,
                                                 (short)0, acc0, false, false);
    acc1 = __builtin_amdgcn_wmma_f32_16x16x4_f32(false, a1, false, b,
                                                 (short)0, acc1, false, false);
  }

  const int col = n0 + mn;
  const float bv = bias ? bias[col] : 0.0f;
  #pragma unroll
  for (int r = 0; r < 8; ++r) {
    int row = r0 + r + 8 * half;        // C layout: VGPR r -> M = r (+8 hi half)
    float x0 = acc0[r] + bv;
    float x1 = acc1[r] + bv;
    if (EPI_RELU) { x0 = fmaxf(x0, 0.0f); x1 = fmaxf(x1, 0.0f); }
    C[(size_t)row * D + col]        = x0;
    C[(size_t)(row + 16) * D + col] = x1;
  }
}

// ---------------------------------------------------------------------------
// out[dst] += w * h[src]  — one wave per edge, 4 floats per lane
// ---------------------------------------------------------------------------
__global__ __launch_bounds__(256) void spmm_atomic(
    const float* __restrict__ h, const float* __restrict__ ew,
    const int* __restrict__ src, const int* __restrict__ dst,
    float* __restrict__ out, int E)
{
  int gid = blockIdx.x * 256 + threadIdx.x;
  int e = gid >> 5;
  if (e >= E) return;
  int lane = threadIdx.x & 31;
  int s = src[e];
  int d = dst[e];
  float w = ew[e];
  const float4 v = *(const float4*)(h + (size_t)s * D + lane * 4);
  float* o = out + (size_t)d * D + lane * 4;
  atomAddF(o + 0, v.x * w);
  atomAddF(o + 1, v.y * w);
  atomAddF(o + 2, v.z * w);
  atomAddF(o + 3, v.w * w);
}

// segment-sum into psum[G x 128] + counts — one wave per node
__global__ __launch_bounds__(256) void pool_accum(
    const float* __restrict__ h, const int* __restrict__ gidv,
    float* __restrict__ psum, float* __restrict__ cnt, int N)
{
  int gid = blockIdx.x * 256 + threadIdx.x;
  int n = gid >> 5;
  if (n >= N) return;
  int lane = threadIdx.x & 31;
  int g = gidv[n];
  const float4 v = *(const float4*)(h + (size_t)n * D + lane * 4);
  float* p = psum + (size_t)g * D + lane * 4;
  atomAddF(p + 0, v.x);
  atomAddF(p + 1, v.y);
  atomAddF(p + 2, v.z);
  atomAddF(p + 3, v.w);
  if (lane == 0) atomAddF(cnt + g, 1.0f);
}

// pooled = psum / max(cnt,1) + b2   (G*D elements)
__global__ __launch_bounds__(256) void pool_finalize(
    const float* __restrict__ psum, const float* __restrict__ cnt,
    const float* __restrict__ b2, float* __restrict__ pooled, int total)
{
  int i = blockIdx.x * 256 + threadIdx.x;
  if (i >= total) return;
  int g = i >> 7;
  int d = i & 127;
  float c = fmaxf(cnt[g], 1.0f);
  pooled[i] = psum[i] / c + b2[d];
}

// S = sigmoid(P3 + P4)   (G*D elements)
__global__ __launch_bounds__(256) void combine_sigmoid(
    const float* __restrict__ P3, const float* __restrict__ P4,
    float* __restrict__ S, int total)
{
  int i = blockIdx.x * 256 + threadIdx.x;
  if (i >= total) return;
  float x = P3[i] + P4[i];
  S[i] = 1.0f / (1.0f + expf(-x));
}

// out[n] = S[graph_id[n]]  — one wave per node, float4 copies
__global__ __launch_bounds__(256) void scatter_out(
    const float* __restrict__ S, const int* __restrict__ gidv,
    float* __restrict__ out, int N)
{
  int gid = blockIdx.x * 256 + threadIdx.x;
  int n = gid >> 5;
  if (n >= N) return;
  int lane = threadIdx.x & 31;
  int g = gidv[n];
  *(float4*)(out + (size_t)n * D + lane * 4) =
      *(const float4*)(S + (size_t)g * D + lane * 4);
}

extern "C" void kernel_launch(void* const* d_in, const int* in_sizes, int n_in,
                              void* d_out, int out_size, void* d_ws, size_t ws_size,
                              hipStream_t stream) {
  const float* feat = (const float*)d_in[0];
  const float* ew   = (const float*)d_in[1];
  const float* W1   = (const float*)d_in[2];
  const float* b1   = (const float*)d_in[3];
  const float* W2   = (const float*)d_in[4];
  const float* b2   = (const float*)d_in[5];
  const float* ffW1 = (const float*)d_in[6];
  const float* ffb1 = (const float*)d_in[7];
  const float* ffW2 = (const float*)d_in[8];
  const float* ffb2 = (const float*)d_in[9];
  const float* ffW3 = (const float*)d_in[10];
  const float* ffb3 = (const float*)d_in[11];
  const float* ffWs = (const float*)d_in[12];
  const float* ffbs = (const float*)d_in[13];
  const int* esrc   = (const int*)d_in[14];
  const int* edst   = (const int*)d_in[15];
  const int* gidv   = (const int*)d_in[16];

  const int N = in_sizes[0] / D;      // 100000
  const int E = in_sizes[1];          // 1600000
  const int GD = G_GRAPHS * D;        // 32768

  // workspace layout: bufA (N*D) then small pooled-level scratch (~1MB)
  float* bufA   = (float*)d_ws;
  float* psum   = bufA + (size_t)N * D;
  float* pooled = psum + GD;
  float* P1     = pooled + GD;
  float* P2     = P1 + GD;
  float* P3     = P2 + GD;
  float* P4     = P3 + GD;
  float* S      = P4 + GD;
  float* cnt    = S + GD;
  float* outp   = (float*)d_out;      // doubles as second N*D buffer

  dim3 blk(256);
  const int gemmN  = N / 32;                  // 3125 blocks
  const int gemmG  = G_GRAPHS / 32;           // 8 blocks
  const int nodeB  = (N * 32 + 255) / 256;    // wave-per-node kernels
  const int edgeB  = (E * 32 + 255) / 256;    // wave-per-edge kernels
  const int gdB    = (GD + 255) / 256;        // 128 blocks

  // 1) bufA = feat @ W1
  gemm128_wmma<false, false><<<gemmN, blk, 0, stream>>>(feat, W1, nullptr,
                                                        nullptr, bufA);
  // 2) out = spmm(bufA)
  hipMemsetAsync(outp, 0, (size_t)N * D * sizeof(float), stream);
  spmm_atomic<<<edgeB, blk, 0, stream>>>(bufA, ew, esrc, edst, outp, E);
  // 3+4) bufA = relu(out + b1) @ W2   (bias+relu fused into A-tile staging)
  gemm128_wmma<true, false><<<gemmN, blk, 0, stream>>>(outp, W2, b1,
                                                       nullptr, bufA);
  // 5) out = spmm(bufA)
  hipMemsetAsync(outp, 0, (size_t)N * D * sizeof(float), stream);
  spmm_atomic<<<edgeB, blk, 0, stream>>>(bufA, ew, esrc, edst, outp, E);
  // 6) pooled = segmean(out) + b2   (mean(x)+b2 == mean(x+b2))
  hipMemsetAsync(psum, 0, (size_t)GD * sizeof(float), stream);
  hipMemsetAsync(cnt, 0, G_GRAPHS * sizeof(float), stream);
  pool_accum<<<nodeB, blk, 0, stream>>>(outp, gidv, psum, cnt, N);
  pool_finalize<<<gdB, blk, 0, stream>>>(psum, cnt, b2, pooled, GD);
  // 7) FF block at pooled (256-row) level only
  gemm128_wmma<false, true><<<gemmG, blk, 0, stream>>>(pooled, ffW1, nullptr,
                                                       ffb1, P1);
  gemm128_wmma<false, true><<<gemmG, blk, 0, stream>>>(P1, ffW2, nullptr,
                                                       ffb2, P2);
  gemm128_wmma<false, true><<<gemmG, blk, 0, stream>>>(P2, ffW3, nullptr,
                                                       ffb3, P3);
  gemm128_wmma<false, false><<<gemmG, blk, 0, stream>>>(pooled, ffWs, nullptr,
                                                        ffbs, P4);
  combine_sigmoid<<<gdB, blk, 0, stream>>>(P3, P4, S, GD);
  // 8) out[n] = S[graph_id[n]]
  scatter_out<<<nodeB, blk, 0, stream>>>(S, gidv, outp, N);
}